// NGCF_28681791602974
// MI455X (gfx1250) — compile-verified
//
#include <hip/hip_runtime.h>

#define N_USERS   100000
#define N_ITEMS   50000
#define N_NODES   150000
#define N_EDGES   4800000
#define DIM       64
#define OUT_STRIDE 256        // 4 * 64 concatenated layer slices
#define LRELU_SLOPE 0.01f
#define NORM_EPS  1e-12f
#define WLD       65          // padded LDS stride for weights

typedef __attribute__((ext_vector_type(2))) float v2f;
typedef __attribute__((ext_vector_type(8))) float v8f;

// ---- write initial ego (concat user/item emb) into output columns [0,64) ----
__global__ __launch_bounds__(256) void k_init_out(const float* __restrict__ ue,
                                                  const float* __restrict__ ie,
                                                  float* __restrict__ out) {
  int idx = blockIdx.x * 256 + threadIdx.x;       // float4 index over [N_NODES][16]
  if (idx >= N_NODES * 16) return;
  int r = idx >> 4, c4 = idx & 15;
  float4 v = (r < N_USERS) ? ((const float4*)ue)[r * 16 + c4]
                           : ((const float4*)ie)[(r - N_USERS) * 16 + c4];
  ((float4*)out)[(size_t)r * (OUT_STRIDE / 4) + c4] = v;
}

// ---- zero the side accumulator ----
__global__ __launch_bounds__(256) void k_zero(float4* __restrict__ p, int n4) {
  int i = blockIdx.x * 256 + threadIdx.x;
  if (i < n4) p[i] = make_float4(0.f, 0.f, 0.f, 0.f);
}

// ---- COO SpMM: side[row] += val * ego[col]   (16 lanes per edge, float4 each)
__global__ __launch_bounds__(256) void k_spmm(const int*   __restrict__ rows,
                                              const int*   __restrict__ cols,
                                              const float* __restrict__ vals,
                                              const float* __restrict__ ego,   // stride OUT_STRIDE
                                              float*       __restrict__ side) {
  unsigned t = blockIdx.x * 256u + threadIdx.x;
  unsigned e = t >> 4;
  int q = t & 15;
  if (e >= N_EDGES) return;
  int   r = rows[e];
  int   c = cols[e];
  float v = vals[e];
  float4 x = *(const float4*)(ego + (size_t)c * OUT_STRIDE + q * 4);
  float* dst = side + (size_t)r * DIM + q * 4;
  unsafeAtomicAdd(dst + 0, v * x.x);
  unsafeAtomicAdd(dst + 1, v * x.y);
  unsafeAtomicAdd(dst + 2, v * x.z);
  unsafeAtomicAdd(dst + 3, v * x.w);
}

// ---- dense layer: ego_out = leaky(side@Wgc + bgc) + leaky((ego*side)@Wbi + bbi)
// one wave per 16-row tile, fp32 WMMA 16x16x4, K=64 in 16 chunks, 4 N-tiles
__global__ __launch_bounds__(256) void k_dense(const float* __restrict__ side,
                                               const float* __restrict__ ego_in,   // stride OUT_STRIDE
                                               const float* __restrict__ Wgc,
                                               const float* __restrict__ Bgc,
                                               const float* __restrict__ Wbi,
                                               const float* __restrict__ Bbi,
                                               float* __restrict__ ego_out) {      // stride OUT_STRIDE
  __shared__ float sWgc[64 * WLD];
  __shared__ float sWbi[64 * WLD];
  __shared__ float sBgc[64];
  __shared__ float sBbi[64];

  for (int i = threadIdx.x; i < 64 * 64; i += 256) {
    int k = i >> 6, n = i & 63;
    sWgc[k * WLD + n] = Wgc[i];
    sWbi[k * WLD + n] = Wbi[i];
  }
  if (threadIdx.x < 64) {
    sBgc[threadIdx.x] = Bgc[threadIdx.x];
    sBbi[threadIdx.x] = Bbi[threadIdx.x];
  }
  __syncthreads();

  int wave = threadIdx.x >> 5;
  int lane = threadIdx.x & 31;
  int tile = blockIdx.x * 8 + wave;
  int row0 = tile * 16;
  if (row0 >= N_NODES) return;            // wave-uniform: EXEC stays all-ones

  int m  = lane & 15;                     // A: row within tile / B: col within N-tile
  int hi = lane >> 4;

  const float* srow = side   + (size_t)(row0 + m) * DIM;
  const float* erow = ego_in + (size_t)(row0 + m) * OUT_STRIDE;

  // A fragments for all 16 K-chunks: element j of lane holds A[m][4*kc + 2*hi + j]
  v2f aS[16], aB[16];
#pragma unroll
  for (int kc = 0; kc < 16; ++kc) {
    int k0 = kc * 4 + hi * 2;
    float s0 = srow[k0], s1 = srow[k0 + 1];
    float e0 = erow[k0], e1 = erow[k0 + 1];
    v2f t0; t0.x = s0;      t0.y = s1;      aS[kc] = t0;
    v2f t1; t1.x = s0 * e0; t1.y = s1 * e1; aB[kc] = t1;
  }

#pragma unroll
  for (int nt = 0; nt < 4; ++nt) {
    v8f accG = {};
    v8f accB = {};
    int nb = nt * 16 + m;                 // column within the 64-wide output
#pragma unroll
    for (int kc = 0; kc < 16; ++kc) {
      int k0 = kc * 4 + hi * 2;
      // B fragment: element j holds W[k0 + j][nb]
      v2f bg; bg.x = sWgc[k0 * WLD + nb]; bg.y = sWgc[(k0 + 1) * WLD + nb];
      v2f bb; bb.x = sWbi[k0 * WLD + nb]; bb.y = sWbi[(k0 + 1) * WLD + nb];
      accG = __builtin_amdgcn_wmma_f32_16x16x4_f32(false, aS[kc], false, bg,
                                                   (short)0, accG, false, false);
      accB = __builtin_amdgcn_wmma_f32_16x16x4_f32(false, aB[kc], false, bb,
                                                   (short)0, accB, false, false);
    }
    float bgc = sBgc[nb], bbi = sBbi[nb];
#pragma unroll
    for (int v = 0; v < 8; ++v) {         // D layout: VGPR v -> row v + 8*hi
      float g = accG[v] + bgc; g = (g > 0.f) ? g : LRELU_SLOPE * g;
      float b = accB[v] + bbi; b = (b > 0.f) ? b : LRELU_SLOPE * b;
      ego_out[(size_t)(row0 + v + 8 * hi) * OUT_STRIDE + nb] = g + b;
    }
  }
}

// ---- in-place L2 normalize output slices 1..3 (slice 0 stays raw, per reference)
__global__ __launch_bounds__(256) void k_norm(float* __restrict__ out) {
  int wave = threadIdx.x >> 5, lane = threadIdx.x & 31;
  int r = blockIdx.x * 8 + wave;
  if (r >= N_NODES) return;
#pragma unroll
  for (int s = 1; s <= 3; ++s) {
    float* p = out + (size_t)r * OUT_STRIDE + s * 64;
    float2 x = *(float2*)(p + lane * 2);
    float sum = x.x * x.x + x.y * x.y;
#pragma unroll
    for (int off = 16; off; off >>= 1) sum += __shfl_xor(sum, off, 32);
    float sc = 1.f / fmaxf(sqrtf(sum), NORM_EPS);
    float2 y; y.x = x.x * sc; y.y = x.y * sc;
    *(float2*)(p + lane * 2) = y;
  }
}

extern "C" void kernel_launch(void* const* d_in, const int* in_sizes, int n_in,
                              void* d_out, int out_size, void* d_ws, size_t ws_size,
                              hipStream_t stream) {
  const int*   rows = (const int*)d_in[0];
  const int*   cols = (const int*)d_in[1];
  const float* vals = (const float*)d_in[2];
  const float* ue   = (const float*)d_in[3];
  const float* ie   = (const float*)d_in[4];
  const float* gcw  = (const float*)d_in[5];
  const float* gcb  = (const float*)d_in[6];
  const float* biw  = (const float*)d_in[7];
  const float* bib  = (const float*)d_in[8];
  float* out  = (float*)d_out;
  float* side = (float*)d_ws;             // 150000*64 f32 = 38.4 MB scratch

  k_init_out<<<(N_NODES * 16 + 255) / 256, 256, 0, stream>>>(ue, ie, out);

  for (int l = 0; l < 3; ++l) {
    k_zero<<<(N_NODES * 16 + 255) / 256, 256, 0, stream>>>((float4*)side, N_NODES * 16);
    k_spmm<<<(N_EDGES * 16) / 256, 256, 0, stream>>>(rows, cols, vals,
                                                     out + l * 64, side);
    k_dense<<<(N_NODES / 16 + 7) / 8, 256, 0, stream>>>(
        side, out + l * 64,
        gcw + l * 4096, gcb + l * 64,
        biw + l * 4096, bib + l * 64,
        out + (l + 1) * 64);
  }

  k_norm<<<N_NODES / 8, 256, 0, stream>>>(out);
}